// MVGlimpseNetwork_38199439131203
// MI455X (gfx1250) — compile-verified
//
#include <hip/hip_runtime.h>
#include <hip/hip_bf16.h>

#define B_    64
#define T_    4096
#define V_    64
#define NG_   64          // NGRAN * NGLIMPSE
#define NHID_ 2048
#define FANIN_ 4097       // V*NG + 1

typedef __attribute__((ext_vector_type(2))) float v2f;
typedef __attribute__((ext_vector_type(8))) float v8f;

// ---------------------------------------------------------------------------
// Phase 1: masked linear interpolation ("glimpses").
// One block per sample b, one thread per variable v. Streaming pass over the
// (sorted) time row; per-granularity query pointers reproduce the reference's
// stable-argsort + searchsorted(side='right') semantics exactly.
// ---------------------------------------------------------------------------
__global__ __launch_bounds__(64)
void glimpse_kernel(const float* __restrict__ vals,   // [B, T, V]
                    const float* __restrict__ time,   // [B, T] sorted asc
                    const int*   __restrict__ masks,  // [B, T, V]
                    const float* __restrict__ l_t,    // [B]
                    float* __restrict__ g_ws,         // [B, V*NG]
                    float* __restrict__ out_mid)      // [B] = g[:, 2048]
{
    __shared__ float s_t[T_];
    __shared__ float s_r[NG_];
    __shared__ float s_res[V_ * NG_];

    const int b = blockIdx.x;
    const int v = threadIdx.x;       // 0..63

    // cooperative, coalesced load of the time row into LDS
    for (int i = v; i < T_; i += V_) s_t[i] = time[b * T_ + i];
    __syncthreads();

    // times are sorted ascending per sample -> max is the last element
    const float tmax = s_t[T_ - 1];
    const float l    = l_t[b];
    {   // 64 threads compute the 64 query times (granularity-major)
        const int   q     = v;
        const int   gran  = q >> 5;
        const int   j     = q & 31;
        const float scale = gran ? 5.0f : 1.0f;
        const float lin   = -0.05f * scale + (float)j * (0.1f * scale / 31.0f);
        s_r[q] = (lin + l) * tmax;
    }
    __syncthreads();

    int   p0 = 0, p1 = 0;            // per-granularity query front pointers
    int   n  = 0;                    // count of valid points seen
    float tp = 0.0f, vp = 0.0f;      // last valid (time, value)
    const long base = (long)b * T_ * V_ + v;
    float* res = &s_res[v * NG_];

    for (int t = 0; t < T_; ++t) {
        const int   m   = masks[base + (long)t * V_];   // coalesced across v
        const float val = vals [base + (long)t * V_];   // coalesced across v
        const float tv  = s_t[t];
        if (m) {
            if (n == 0) {
                // r <= first valid time -> constant fill with first value
                while (p0 < 32 && s_r[p0]      <= tv) res[p0++]      = val;
                while (p1 < 32 && s_r[32 + p1] <= tv) res[32 + p1++] = val;
            } else {
                const float dt = (tv > tp) ? (tv - tp) : 1.0f;
                while (p0 < 32) {
                    const float r = s_r[p0];
                    if (r > tv) break;
                    float w = (r - tp) / dt;
                    w = fminf(fmaxf(w, 0.0f), 1.0f);
                    res[p0++] = vp + w * (val - vp);
                }
                while (p1 < 32) {
                    const float r = s_r[32 + p1];
                    if (r > tv) break;
                    float w = (r - tp) / dt;
                    w = fminf(fmaxf(w, 0.0f), 1.0f);
                    res[32 + p1++] = vp + w * (val - vp);
                }
            }
            tp = tv; vp = val; ++n;
        }
    }
    // r beyond last valid time -> constant fill with last value; 0 if none
    const float fill = n ? vp : 0.0f;
    while (p0 < 32) res[p0++]      = fill;
    while (p1 < 32) res[32 + p1++] = fill;
    __syncthreads();

    // coalesced writeback; [v][q] flattening matches reference reshape order
    for (int i = v; i < V_ * NG_; i += V_)
        g_ws[(long)b * (V_ * NG_) + i] = s_res[i];

    if (v == 0) out_mid[b] = s_res[32 * NG_ + 0];   // g[:, 2048]
}

// ---------------------------------------------------------------------------
// Phase 2: grep = [g | l_t] @ W^T + bias via V_WMMA_F32_16X16X4_F32.
// One wave per 16x16 output tile; 512 tiles total. A (16x4 f32): lanes 0-15
// carry K=0,1 in v[0],v[1]; lanes 16-31 carry K=2,3 (per ISA layout), so each
// lane issues one b64 load per K-step from row-major X. B (4x16) is mirrored
// from W's K-contiguous rows. The l_t column (k=4096) and bias are folded in
// with a scalar FMA at the end.
// ---------------------------------------------------------------------------
__global__ __launch_bounds__(256)
void gemm_wmma_kernel(const float* __restrict__ g,     // [64, 4096]
                      const float* __restrict__ l_t,   // [64]
                      const float* __restrict__ W,     // [2048, 4097]
                      const float* __restrict__ bias,  // [2048]
                      float* __restrict__ out)         // [64, 2048]
{
    const int lane = threadIdx.x & 31;
    const int wave = threadIdx.x >> 5;
    const int tile = blockIdx.x * 8 + wave;  // 0..511
    const int m0   = (tile & 3) * 16;        // 4 M-tiles
    const int n0   = (tile >> 2) * 16;       // 128 N-tiles
    const int half = lane >> 4;
    const int l15  = lane & 15;

    const float* Ap = g + (long)(m0 + l15) * 4096  + 2 * half;
    const float* Bp = W + (long)(n0 + l15) * FANIN_ + 2 * half;

    v8f c = {0.f, 0.f, 0.f, 0.f, 0.f, 0.f, 0.f, 0.f};
    for (int k = 0; k < 4096; k += 4) {
        v2f a  = *(const v2f*)(Ap + k);
        v2f bb = *(const v2f*)(Bp + k);
        c = __builtin_amdgcn_wmma_f32_16x16x4_f32(
                /*neg_a=*/false, a, /*neg_b=*/false, bb,
                /*c_mod=*/(short)0, c, /*reuse_a=*/false, /*reuse_b=*/false);
    }

    const float wlast = W[(long)(n0 + l15) * FANIN_ + 4096];  // l_t column
    const float bn    = bias[n0 + l15];
#pragma unroll
    for (int r = 0; r < 8; ++r) {
        const int m = m0 + r + 8 * half;   // C/D layout: VGPR r -> M=r(+8)
        out[(long)m * NHID_ + n0 + l15] = c[r] + bn + l_t[m] * wlast;
    }
}

// ---------------------------------------------------------------------------
extern "C" void kernel_launch(void* const* d_in, const int* in_sizes, int n_in,
                              void* d_out, int out_size, void* d_ws, size_t ws_size,
                              hipStream_t stream) {
    const float* vals  = (const float*)d_in[0];   // [64,4096,64] f32
    const float* time  = (const float*)d_in[1];   // [64,4096]    f32
    const int*   masks = (const int*)  d_in[2];   // [64,4096,64] i32
    // d_in[3] = lengths (unused by the reference computation)
    const float* l_t   = (const float*)d_in[4];   // [64,1]       f32
    const float* W     = (const float*)d_in[5];   // [2048,4097]  f32
    const float* bias  = (const float*)d_in[6];   // [2048]       f32

    float* grep    = (float*)d_out;               // [64,2048]
    float* out_mid = (float*)d_out + B_ * NHID_;  // [64]
    float* g_ws    = (float*)d_ws;                // [64,4096] scratch (1 MB)

    glimpse_kernel<<<B_, V_, 0, stream>>>(vals, time, masks, l_t, g_ws, out_mid);
    gemm_wmma_kernel<<<64, 256, 0, stream>>>(g_ws, l_t, W, bias, grep);
}